// ReviewAggregatorConv_14525579395328
// MI455X (gfx1250) — compile-verified
//
#include <hip/hip_runtime.h>
#include <hip/hip_bf16.h>
#include <math.h>

typedef float v2f __attribute__((ext_vector_type(2)));
typedef float v8f __attribute__((ext_vector_type(8)));

#define IN_DIM  128
#define ATT_DIM 64

// ---------------------------------------------------------------- utilities

__global__ void fill_kernel(float* __restrict__ p, int n, float v) {
    int i = blockIdx.x * blockDim.x + threadIdx.x;
    if (i < n) p[i] = v;
}

__device__ inline void atomicMaxF(float* addr, float v) {
    // Standard IEEE-ordering trick: int-max for >=0, uint-min for <0.
    if (v >= 0.0f) atomicMax((int*)addr, __float_as_int(v));
    else           atomicMin((unsigned int*)addr, __float_as_uint(v));
}

// ------------------------------------------------- column sum of X [M,128]

__global__ void colsum_kernel(const float* __restrict__ X, int M,
                              float* __restrict__ out) {
    const int c  = threadIdx.x;              // 128 threads, one per column
    int r0 = blockIdx.x * 512;
    int r1 = r0 + 512; if (r1 > M) r1 = M;
    float acc = 0.0f;
    for (int r = r0; r < r1; ++r) acc += X[(size_t)r * IN_DIM + c];
    atomicAdd(&out[c], acc);
}

// g[n] = bg[n] + colsum[k] * Wg[k,n]  (tiny: 1 block, 64 threads)
__global__ void gvec_kernel(const float* __restrict__ colsum,
                            const float* __restrict__ Wg,
                            const float* __restrict__ bg,
                            float* __restrict__ g) {
    const int n = threadIdx.x;               // 64 threads
    float acc = bg[n];
    #pragma unroll 4
    for (int k = 0; k < IN_DIM; ++k) acc += colsum[k] * Wg[k * ATT_DIM + n];
    g[n] = acc;
}

// ------------------------------------------------------------- WMMA GEMM
// out[M,64] = X[M,128] @ W[128,64] + bias[64] (+ grow[64] if non-null)
// One block = 4 waves; each wave owns one 16-wide N tile; block owns 16 rows.
// M must be a multiple of 16 (100000 and 50000 both are).

__launch_bounds__(128)
__global__ void gemm_xw_kernel(const float* __restrict__ X,
                               const float* __restrict__ W,
                               const float* __restrict__ bias,
                               const float* __restrict__ grow,
                               float* __restrict__ out, int M) {
    __shared__ float sA[16 * IN_DIM];        // 8 KB  : 16-row X tile
    __shared__ float sW[IN_DIM * ATT_DIM];   // 32 KB : full weight matrix

    const int tid   = threadIdx.x;
    const int tileM = blockIdx.x;

    const float* Xrow = X + (size_t)tileM * 16 * IN_DIM;
    for (int i = tid; i < 16 * IN_DIM; i += 128)       sA[i] = Xrow[i];
    for (int i = tid; i < IN_DIM * ATT_DIM; i += 128)  sW[i] = W[i];
    __syncthreads();

    const int wave  = tid >> 5;              // 0..3 -> N tile
    const int lane  = tid & 31;
    const int laneM = lane & 15;             // A: row within tile
    const int koff  = (lane >> 4) * 2;       // A/B: K sub-offset per half-wave
    const int ncol  = wave * 16 + (lane & 15); // B/D: output column

    v8f c = {};
    #pragma unroll 8
    for (int kk = 0; kk < IN_DIM / 4; ++kk) {
        const int k0 = kk * 4 + koff;
        v2f a, b;
        a.x = sA[laneM * IN_DIM + k0];
        a.y = sA[laneM * IN_DIM + k0 + 1];
        b.x = sW[k0 * ATT_DIM + ncol];
        b.y = sW[(k0 + 1) * ATT_DIM + ncol];
        c = __builtin_amdgcn_wmma_f32_16x16x4_f32(
                /*neg_a=*/false, a, /*neg_b=*/false, b,
                /*c_mod=*/(short)0, c, /*reuse_a=*/false, /*reuse_b=*/false);
    }

    const float add = bias[ncol] + (grow ? grow[ncol] : 0.0f);

    // D layout: lanes 0-15 N=lane, lanes 16-31 N=lane-16; VGPR v -> M=v(+8)
    const int mbase   = (lane >> 4) * 8;
    const size_t rowb = (size_t)tileM * 16;
    #pragma unroll
    for (int v = 0; v < 8; ++v)
        out[(rowb + mbase + v) * ATT_DIM + ncol] = c[v] + add;
}

// --------------------------------------------------------- edge attention
// s[e] = leaky_relu(Hsrc[src[e]] + Hnp[npid[e]]) . Wa + ba ; atomicMax dmax[dst]
// One wave (32 lanes) per edge; each lane covers dims {lane, lane+32}.

__launch_bounds__(256)
__global__ void edge_score_kernel(const float* __restrict__ Hsrc,
                                  const float* __restrict__ Hnp,
                                  const int* __restrict__ src,
                                  const int* __restrict__ npid,
                                  const int* __restrict__ dst,
                                  const float* __restrict__ Wa,
                                  const float* __restrict__ ba,
                                  float* __restrict__ s,
                                  float* __restrict__ dmax, int E) {
    const int wave = threadIdx.x >> 5;
    const int lane = threadIdx.x & 31;
    const int e = blockIdx.x * 8 + wave;
    if (e >= E) return;

    const int si = src[e], ni = npid[e];
    const float* hp = Hsrc + (size_t)si * ATT_DIM;
    const float* np = Hnp  + (size_t)ni * ATT_DIM;

    float t0 = hp[lane]      + np[lane];
    float t1 = hp[lane + 32] + np[lane + 32];
    t0 = (t0 >= 0.0f) ? t0 : 0.01f * t0;    // LeakyReLU(0.01)
    t1 = (t1 >= 0.0f) ? t1 : 0.01f * t1;
    float v = t0 * Wa[lane] + t1 * Wa[lane + 32];

    #pragma unroll
    for (int off = 16; off > 0; off >>= 1)
        v += __shfl_xor(v, off, 32);

    if (lane == 0) {
        const float sv = v + ba[0];
        s[e] = sv;
        atomicMaxF(&dmax[dst[e]], sv);
    }
}

// s[e] <- exp(s[e] - dmax[dst]) ; z[dst] += s[e]
__global__ void edge_exp_kernel(float* __restrict__ s,
                                const int* __restrict__ dst,
                                const float* __restrict__ dmax,
                                float* __restrict__ z, int E) {
    const int e = blockIdx.x * blockDim.x + threadIdx.x;
    if (e >= E) return;
    const int d = dst[e];
    const float v = expf(s[e] - dmax[d]);
    s[e] = v;
    atomicAdd(&z[d], v);
}

// out[dst] += x[src] * (s[e]/z[dst]) ; wave per edge, float4 per lane.
__launch_bounds__(256)
__global__ void edge_scatter_kernel(const float* __restrict__ s,
                                    const float* __restrict__ z,
                                    const float* __restrict__ Xsrc,
                                    const int* __restrict__ src,
                                    const int* __restrict__ dst,
                                    float* __restrict__ out, int E) {
    const int wave = threadIdx.x >> 5;
    const int lane = threadIdx.x & 31;
    const int e = blockIdx.x * 8 + wave;
    if (e >= E) return;

    const int si = src[e], d = dst[e];
    const float a = s[e] / z[d];

    const float4 xv = ((const float4*)(Xsrc + (size_t)si * IN_DIM))[lane];
    float* op = out + (size_t)d * IN_DIM + lane * 4;
    atomicAdd(op + 0, xv.x * a);
    atomicAdd(op + 1, xv.y * a);
    atomicAdd(op + 2, xv.z * a);
    atomicAdd(op + 3, xv.w * a);
}

// ---------------------------------------------------------------- launcher

extern "C" void kernel_launch(void* const* d_in, const int* in_sizes, int n_in,
                              void* d_out, int out_size, void* d_ws, size_t ws_size,
                              hipStream_t stream) {
    const float* x_user   = (const float*)d_in[0];
    const float* x_item   = (const float*)d_in[1];
    const float* Wo       = (const float*)d_in[2];
    const float* bo       = (const float*)d_in[3];
    const float* Wu       = (const float*)d_in[4];
    const float* bu       = (const float*)d_in[5];
    const float* Wg       = (const float*)d_in[6];
    const float* bg       = (const float*)d_in[7];
    const float* Wa       = (const float*)d_in[8];
    const float* ba       = (const float*)d_in[9];
    const float* emb_user = (const float*)d_in[10];
    const float* emb_item = (const float*)d_in[11];
    const int* src_iu  = (const int*)d_in[12];
    const int* dst_iu  = (const int*)d_in[13];
    const int* npid_iu = (const int*)d_in[14];
    const int* src_ui  = (const int*)d_in[15];
    const int* dst_ui  = (const int*)d_in[16];
    const int* npid_ui = (const int*)d_in[17];

    const int NU = in_sizes[0] / IN_DIM;     // 100000
    const int NI = in_sizes[1] / IN_DIM;     // 50000
    const int E  = in_sizes[12];             // 600000

    float* out = (float*)d_out;

    // ---- workspace layout (floats) ----
    float* w = (float*)d_ws;
    float* h_user  = w;  w += (size_t)NU * ATT_DIM;
    float* h_item  = w;  w += (size_t)NI * ATT_DIM;
    float* hu_user = w;  w += (size_t)NU * ATT_DIM;
    float* hu_item = w;  w += (size_t)NI * ATT_DIM;
    float* s_iu    = w;  w += E;
    float* s_ui    = w;  w += E;
    float* dmax_u  = w;  w += NU;            // dmax_u/dmax_i contiguous
    float* dmax_i  = w;  w += NI;
    float* z_u     = w;  w += NU;            // z_u/z_i contiguous
    float* z_i     = w;  w += NI;
    float* cs_u    = w;  w += IN_DIM;        // cs_u/cs_i contiguous
    float* cs_i    = w;  w += IN_DIM;
    float* g_u     = w;  w += ATT_DIM;
    float* g_i     = w;  w += ATT_DIM;
    (void)g_i; (void)ws_size; (void)n_in; (void)out_size;

    // ---- init: zero output, zero accumulators, -inf maxes ----
    {
        const int nOut = (NU + NI) * IN_DIM;
        fill_kernel<<<(nOut + 255) / 256, 256, 0, stream>>>(out, nOut, 0.0f);
        fill_kernel<<<(NU + NI + 255) / 256, 256, 0, stream>>>(dmax_u, NU + NI,
                                                               -__builtin_huge_valf());
        fill_kernel<<<(NU + NI + 255) / 256, 256, 0, stream>>>(z_u, NU + NI, 0.0f);
        fill_kernel<<<1, 256, 0, stream>>>(cs_u, 2 * IN_DIM, 0.0f);
    }

    // ---- column sums + g vectors ----
    colsum_kernel<<<(NU + 511) / 512, IN_DIM, 0, stream>>>(x_user, NU, cs_u);
    colsum_kernel<<<(NI + 511) / 512, IN_DIM, 0, stream>>>(x_item, NI, cs_i);
    gvec_kernel<<<1, ATT_DIM, 0, stream>>>(cs_u, Wg, bg, g_u);
    gvec_kernel<<<1, ATT_DIM, 0, stream>>>(cs_i, Wg, bg, g_i);

    // ---- four WMMA GEMMs (M multiples of 16) ----
    gemm_xw_kernel<<<NU / 16, 128, 0, stream>>>(x_user,   Wo, bo, g_u,     h_user,  NU);
    gemm_xw_kernel<<<NI / 16, 128, 0, stream>>>(x_item,   Wo, bo, g_i,     h_item,  NI);
    gemm_xw_kernel<<<NU / 16, 128, 0, stream>>>(emb_user, Wu, bu, nullptr, hu_user, NU);
    gemm_xw_kernel<<<NI / 16, 128, 0, stream>>>(emb_item, Wu, bu, nullptr, hu_item, NI);

    const int gridE8 = (E + 7) / 8;

    // ---- edge type: item -> user ----
    edge_score_kernel<<<gridE8, 256, 0, stream>>>(h_item, hu_item, src_iu, npid_iu,
                                                  dst_iu, Wa, ba, s_iu, dmax_u, E);
    edge_exp_kernel<<<(E + 255) / 256, 256, 0, stream>>>(s_iu, dst_iu, dmax_u, z_u, E);
    edge_scatter_kernel<<<gridE8, 256, 0, stream>>>(s_iu, z_u, x_item, src_iu, dst_iu,
                                                    out, E);

    // ---- edge type: user -> item ----
    edge_score_kernel<<<gridE8, 256, 0, stream>>>(h_user, hu_user, src_ui, npid_ui,
                                                  dst_ui, Wa, ba, s_ui, dmax_i, E);
    edge_exp_kernel<<<(E + 255) / 256, 256, 0, stream>>>(s_ui, dst_ui, dmax_i, z_i, E);
    edge_scatter_kernel<<<gridE8, 256, 0, stream>>>(s_ui, z_i, x_user, src_ui, dst_ui,
                                                    out + (size_t)NU * IN_DIM, E);
}